// Cross_Attention_13116830122231
// MI455X (gfx1250) — compile-verified
//
#include <hip/hip_runtime.h>
#include <hip/hip_bf16.h>

// Problem constants (B=8, N=4096, D=512, 2D=1024)
#define PB 8
#define PN 4096
#define PD 512
#define PD2 1024

typedef float v2f __attribute__((ext_vector_type(2)));
typedef float v4f __attribute__((ext_vector_type(4)));
typedef float v8f __attribute__((ext_vector_type(8)));

// CDNA5 fp32 WMMA: D(16x16,f32) = A(16x4,f32) * B(4x16,f32) + C
__device__ __forceinline__ v8f wmma4(v2f a, v2f b, v8f c) {
    return __builtin_amdgcn_wmma_f32_16x16x4_f32(
        /*neg_a=*/false, a, /*neg_b=*/false, b,
        /*c_mod=*/(short)0, c, /*reuse_a=*/false, /*reuse_b=*/false);
}

// ---------------------------------------------------------------------------
// Softmax over N (per (b,d) row of keys == x2^T): split reduction.
// Pass 1: partial max & sum(exp(x-m_partial)) per 128-token chunk.
__global__ __launch_bounds__(512) void kstats_partial(const float* __restrict__ x2,
                                                      float* __restrict__ pmax,
                                                      float* __restrict__ psum) {
    int b = blockIdx.y;
    int chunk = blockIdx.x;              // 32 chunks of 128 tokens
    int d = threadIdx.x;                 // 512
    const float* p = x2 + (size_t)b * PN * PD + (size_t)chunk * 128 * PD + d;
    float m = -3.402823466e38f;
    for (int i = 0; i < 128; ++i) m = fmaxf(m, p[(size_t)i * PD]);
    float s = 0.f;
    for (int i = 0; i < 128; ++i) s += __expf(p[(size_t)i * PD] - m);
    int idx = (b * 32 + chunk) * PD + d;
    pmax[idx] = m;
    psum[idx] = s;
}

__global__ __launch_bounds__(512) void kstats_combine(const float* __restrict__ pmax,
                                                      const float* __restrict__ psum,
                                                      float* __restrict__ kmax,
                                                      float* __restrict__ ksum) {
    int b = blockIdx.x;
    int d = threadIdx.x;
    float m = -3.402823466e38f;
    for (int c = 0; c < 32; ++c) m = fmaxf(m, pmax[(b * 32 + c) * PD + d]);
    float s = 0.f;
    for (int c = 0; c < 32; ++c) s += psum[(b * 32 + c) * PD + d] * __expf(pmax[(b * 32 + c) * PD + d] - m);
    kmax[b * PD + d] = m;
    ksum[b * PD + d] = s;
}

// Softmax over D (per (b,n) token): one wave32 per token.
__global__ __launch_bounds__(256) void qstats(const float* __restrict__ x2,
                                              float* __restrict__ qmax,
                                              float* __restrict__ qsum) {
    int tk = blockIdx.x * 8 + (threadIdx.x >> 5);   // flat (b*N+n), 0..32767
    int lane = threadIdx.x & 31;
    const float* p = x2 + (size_t)tk * PD;
    float m = -3.402823466e38f;
#pragma unroll
    for (int i = 0; i < 16; ++i) m = fmaxf(m, p[lane + i * 32]);
#pragma unroll
    for (int off = 16; off > 0; off >>= 1) m = fmaxf(m, __shfl_xor(m, off, 32));
    float s = 0.f;
#pragma unroll
    for (int i = 0; i < 16; ++i) s += __expf(p[lane + i * 32] - m);
#pragma unroll
    for (int off = 16; off > 0; off >>= 1) s += __shfl_xor(s, off, 32);
    if (lane == 0) { qmax[tk] = m; qsum[tk] = s; }
}

// Normalize + transpose x2[b,n,d] -> dst[b,d,n] via 32x32 LDS tile.
// statPerD=1: stats indexed by (b,d) [key softmax]; 0: by (b,n) [query softmax]
__global__ __launch_bounds__(256) void transpose_norm(const float* __restrict__ x2,
                                                      const float* __restrict__ smax,
                                                      const float* __restrict__ ssum,
                                                      float* __restrict__ dst,
                                                      int statPerD) {
    __shared__ float tile[32][33];
    int b = blockIdx.z;
    int n0 = blockIdx.x * 32;
    int d0 = blockIdx.y * 32;
    int tx = threadIdx.x & 31;
    int ty = threadIdx.x >> 5;           // 8 rows per pass
#pragma unroll
    for (int r = 0; r < 32; r += 8) {
        int n = n0 + ty + r;
        int d = d0 + tx;
        float v = x2[((size_t)b * PN + n) * PD + d];
        float m, s;
        if (statPerD) { m = smax[b * PD + d]; s = ssum[b * PD + d]; }
        else          { m = smax[b * PN + n]; s = ssum[b * PN + n]; }
        tile[ty + r][tx] = __expf(v - m) * (1.0f / s);
    }
    __syncthreads();
#pragma unroll
    for (int r = 0; r < 32; r += 8) {
        int d = d0 + ty + r;
        int n = n0 + tx;
        dst[((size_t)b * PD + d) * PN + n] = tile[tx][ty + r];
    }
}

// ---------------------------------------------------------------------------
// GEMM1: context[b] (512x512) = key_sm[b] (512x4096) @ x1[b] (4096x512)
// Wave computes a 16x64 tile (4 N-subtiles), K=4096 in steps of 4.
__global__ __launch_bounds__(256) void gemm1_context(const float* __restrict__ key_sm,
                                                     const float* __restrict__ x1,
                                                     float* __restrict__ ctx) {
    int b = blockIdx.y;
    int task = blockIdx.x * 8 + (threadIdx.x >> 5);  // 0..255
    int mt = task >> 3;                              // 0..31 (16-row tile)
    int ng = task & 7;                               // 0..7  (64-col group)
    int lane = threadIdx.x & 31;
    int m = lane & 15;
    int half = lane >> 4;
    int kh = half * 2;
    int col = lane & 15;
    const float* A  = key_sm + ((size_t)b * PD + mt * 16 + m) * PN + kh;
    const float* Bb = x1 + (size_t)b * PN * PD + (size_t)kh * PD + ng * 64 + col;
    v8f c0 = {}, c1 = {}, c2 = {}, c3 = {};
    for (int k = 0; k < PN; k += 4) {
        const float* pb = Bb + (size_t)k * PD;
        __builtin_prefetch(pb + 32 * PD, 0, 1);
        v2f a;  a.x = A[k];       a.y = A[k + 1];
        v2f b0; b0.x = pb[0];     b0.y = pb[PD];
        v2f b1; b1.x = pb[16];    b1.y = pb[PD + 16];
        v2f b2; b2.x = pb[32];    b2.y = pb[PD + 32];
        v2f b3; b3.x = pb[48];    b3.y = pb[PD + 48];
        c0 = wmma4(a, b0, c0);
        c1 = wmma4(a, b1, c1);
        c2 = wmma4(a, b2, c2);
        c3 = wmma4(a, b3, c3);
    }
    float* o = ctx + ((size_t)b * PD + mt * 16) * PD + ng * 64;
#pragma unroll
    for (int r = 0; r < 8; ++r) {
        int row = r + half * 8;
        o[row * PD + col]      = c0[r];
        o[row * PD + 16 + col] = c1[r];
        o[row * PD + 32 + col] = c2[r];
        o[row * PD + 48 + col] = c3[r];
    }
}

// ---------------------------------------------------------------------------
// Fused 4x top-k masked softmax mix, in place on context rows.
// A[e] = exp(c_e - max) * sum_i (c_e >= t_i) * w_i / s_i
__global__ __launch_bounds__(256) void topk_mix(float* __restrict__ ctx,
                                                const float* __restrict__ attw) {
    __shared__ float orig[512];
    __shared__ float srt[512];
    __shared__ float4 red[256];
    int tid = threadIdx.x;
    float* p = ctx + (size_t)blockIdx.x * PD;   // one (b,d) row
    orig[tid] = p[tid];
    orig[tid + 256] = p[tid + 256];
    srt[tid] = orig[tid];
    srt[tid + 256] = orig[tid + 256];
    // bitonic sort ascending, 512 elements, 256 threads (2 indices each)
    for (int k = 2; k <= 512; k <<= 1) {
        for (int j = k >> 1; j > 0; j >>= 1) {
            __syncthreads();
#pragma unroll
            for (int base = 0; base < 512; base += 256) {
                int i = base + tid;
                int ixj = i ^ j;
                if (ixj > i) {
                    float a = srt[i], b2 = srt[ixj];
                    bool up = ((i & k) == 0);
                    if (up ? (a > b2) : (a < b2)) { srt[i] = b2; srt[ixj] = a; }
                }
            }
        }
    }
    __syncthreads();
    // thresholds: sorted[D-k] for k in {256, 341, 384, 409}
    float t0 = srt[256], t1 = srt[171], t2 = srt[128], t3 = srt[103];
    float mx = srt[511];
    float4 part = {0.f, 0.f, 0.f, 0.f};
    for (int i = tid; i < 512; i += 256) {
        float v = orig[i];
        float e = __expf(v - mx);
        if (v >= t0) part.x += e;
        if (v >= t1) part.y += e;
        if (v >= t2) part.z += e;
        if (v >= t3) part.w += e;
    }
    red[tid] = part;
    __syncthreads();
    for (int s = 128; s > 0; s >>= 1) {
        if (tid < s) {
            red[tid].x += red[tid + s].x; red[tid].y += red[tid + s].y;
            red[tid].z += red[tid + s].z; red[tid].w += red[tid + s].w;
        }
        __syncthreads();
    }
    float w0 = attw[0] / red[0].x, w1 = attw[1] / red[0].y;
    float w2 = attw[2] / red[0].z, w3 = attw[3] / red[0].w;
    for (int i = tid; i < 512; i += 256) {
        float v = orig[i];
        float e = __expf(v - mx);
        float cf = 0.f;
        if (v >= t0) cf += w0;
        if (v >= t1) cf += w1;
        if (v >= t2) cf += w2;
        if (v >= t3) cf += w3;
        p[i] = e * cf;
    }
}

// ---------------------------------------------------------------------------
// GEMM2: agg[b] (512x4096) = A[b] (512x512) @ query_sm[b] (512x4096),
// stored transposed as aggT[b, n, d] so GEMM3 reads K-contiguous rows.
__global__ __launch_bounds__(256) void gemm2_agg(const float* __restrict__ Amix,
                                                 const float* __restrict__ qsm,
                                                 float* __restrict__ aggT) {
    int b = blockIdx.y;
    int task = blockIdx.x * 8 + (threadIdx.x >> 5);  // 0..1023
    int mt = task >> 5;                              // 0..31
    int ng = task & 31;                              // 0..31 (128-col group)
    int lane = threadIdx.x & 31;
    int m = lane & 15;
    int half = lane >> 4;
    int kh = half * 2;
    int col = lane & 15;
    const float* A  = Amix + ((size_t)b * PD + mt * 16 + m) * PD + kh;
    const float* Bb = qsm + (size_t)b * PD * PN + (size_t)kh * PN + ng * 128 + col;
    v8f c[8] = {};
    for (int k = 0; k < PD; k += 4) {
        v2f a; a.x = A[k]; a.y = A[k + 1];
        const float* pb = Bb + (size_t)k * PN;
#pragma unroll
        for (int t = 0; t < 8; ++t) {
            v2f bv; bv.x = pb[t * 16]; bv.y = pb[PN + t * 16];
            c[t] = wmma4(a, bv, c[t]);
        }
    }
#pragma unroll
    for (int t = 0; t < 8; ++t) {
        int n = ng * 128 + t * 16 + col;
        float* pd = aggT + ((size_t)b * PN + n) * PD + mt * 16 + half * 8;
        *(v4f*)pd = c[t].lo;
        *(v4f*)(pd + 4) = c[t].hi;
    }
}

// ---------------------------------------------------------------------------
// GEMM3 + bias + LayerNorm fused.
// Block: 16 tokens x all 1024 outputs. Wave w -> 128-col slab. Then LN via LDS.
__global__ __launch_bounds__(256) void gemm3_ln(const float* __restrict__ aggT,
                                                const float* __restrict__ Wp,
                                                const float* __restrict__ bp,
                                                const float* __restrict__ lnw,
                                                const float* __restrict__ lnb,
                                                float* __restrict__ out) {
    __shared__ float proj[16 * PD2];   // 64 KB of the 320 KB WGP LDS
    int b = blockIdx.y;
    int n0 = blockIdx.x * 16;
    int w = threadIdx.x >> 5;
    int lane = threadIdx.x & 31;
    int m = lane & 15;
    int half = lane >> 4;
    int kh = half * 2;
    int col = lane & 15;
    int o0 = w * 128;
    const float* A = aggT + ((size_t)b * PN + n0 + m) * PD + kh;
    v8f c[8] = {};
    for (int k = 0; k < PD; k += 4) {
        v2f a; a.x = A[k]; a.y = A[k + 1];
#pragma unroll
        for (int t = 0; t < 8; ++t) {
            const float* pw = Wp + (size_t)(o0 + t * 16 + col) * PD + k + kh;
            v2f bv; bv.x = pw[0]; bv.y = pw[1];
            c[t] = wmma4(a, bv, c[t]);
        }
    }
#pragma unroll
    for (int t = 0; t < 8; ++t) {
        int o = o0 + t * 16 + col;
        float bias = bp[o];
#pragma unroll
        for (int r = 0; r < 8; ++r) {
            int row = r + half * 8;
            proj[row * PD2 + o] = c[t][r] + bias;
        }
    }
    __syncthreads();
    // LayerNorm over 1024 channels; wave w handles token rows 2w, 2w+1
#pragma unroll
    for (int rr = 0; rr < 2; ++rr) {
        int row = w * 2 + rr;
        float s = 0.f, s2 = 0.f;
        for (int i = lane; i < PD2; i += 32) {
            float v = proj[row * PD2 + i];
            s += v; s2 += v * v;
        }
#pragma unroll
        for (int off = 16; off > 0; off >>= 1) {
            s  += __shfl_xor(s,  off, 32);
            s2 += __shfl_xor(s2, off, 32);
        }
        float mean = s * (1.0f / PD2);
        float var  = s2 * (1.0f / PD2) - mean * mean;
        float rinv = rsqrtf(var + 1e-5f);
        float* po = out + ((size_t)b * PN + n0 + row) * PD2;
        for (int i = lane; i < PD2; i += 32) {
            float v = proj[row * PD2 + i];
            po[i] = (v - mean) * rinv * lnw[i] + lnb[i];
        }
    }
}

// ---------------------------------------------------------------------------
extern "C" void kernel_launch(void* const* d_in, const int* in_sizes, int n_in,
                              void* d_out, int out_size, void* d_ws, size_t ws_size,
                              hipStream_t stream) {
    (void)in_sizes; (void)n_in; (void)out_size; (void)ws_size;
    const float* x1   = (const float*)d_in[0];
    const float* x2   = (const float*)d_in[1];
    const float* Wp   = (const float*)d_in[2];
    const float* bp   = (const float*)d_in[3];
    const float* lnw  = (const float*)d_in[4];
    const float* lnb  = (const float*)d_in[5];
    const float* attw = (const float*)d_in[6];
    float* out = (float*)d_out;

    char* ws = (char*)d_ws;
    float* key_sm = (float*)(ws);                    // 64 MB; reused as aggT
    float* qsm    = (float*)(ws + (size_t)67108864); // 64 MB
    float* ctx    = (float*)(ws + (size_t)134217728);// 8 MB (context -> A in place)
    float* kmax   = (float*)(ws + (size_t)142606336);
    float* ksum   = kmax + PB * PD;
    float* qmax   = ksum + PB * PD;
    float* qsum   = qmax + PB * PN;
    float* pmax   = qsum + PB * PN;                  // 8*32*512
    float* psum   = pmax + PB * 32 * PD;

    kstats_partial<<<dim3(32, PB), 512, 0, stream>>>(x2, pmax, psum);
    kstats_combine<<<PB, 512, 0, stream>>>(pmax, psum, kmax, ksum);
    qstats<<<PB * PN / 8, 256, 0, stream>>>(x2, qmax, qsum);
    transpose_norm<<<dim3(PN / 32, PD / 32, PB), 256, 0, stream>>>(x2, kmax, ksum, key_sm, 1);
    transpose_norm<<<dim3(PN / 32, PD / 32, PB), 256, 0, stream>>>(x2, qmax, qsum, qsm, 0);
    gemm1_context<<<dim3(32, PB), 256, 0, stream>>>(key_sm, x1, ctx);
    topk_mix<<<PB * PD, 256, 0, stream>>>(ctx, attw);
    gemm2_agg<<<dim3(128, PB), 256, 0, stream>>>(ctx, qsm, /*aggT=*/key_sm);
    gemm3_ln<<<dim3(PN / 16, PB), 256, 0, stream>>>(/*aggT=*/key_sm, Wp, bp, lnw, lnb, out);
}